// attention_module_48885317763455
// MI455X (gfx1250) — compile-verified
//
#include <hip/hip_runtime.h>

// CBAM spatial attention gate for x:[32,256,64,64] f32.
// Memory-bound: >=256MiB HBM traffic vs ~130 MFLOP. Two passes:
//   1) channel max+mean -> comb[B][2][64][64] (1 MiB, stays in L2)
//   2) 7x7 conv on comb via V_WMMA_F32_16X16X4_F32 (weights replicated across
//      all 16 A-rows so every lane's D[0] = conv[lane%16]), sigmoid, then
//      coalesced gated copy x -> out.

typedef __attribute__((ext_vector_type(2))) float v2f;
typedef __attribute__((ext_vector_type(8))) float v8f;

#define NC   256
#define HW   4096
#define WDIM 64

// ---------------- Pass 1: per-pixel channel max + mean ----------------
// One thread = 4 consecutive pixels (float4). Wave reads 512B contiguous per
// channel step; 256-deep channel loop unrolled for memory-level parallelism.
__global__ __launch_bounds__(256) void reduce_maxmean_kernel(
    const float* __restrict__ x, float* __restrict__ comb) {
  int t    = blockIdx.x * 256 + threadIdx.x;   // 0..32767
  int pix0 = t << 2;                           // 4 pixels per thread
  int b    = pix0 >> 12;                       // HW = 4096 pixels/image
  int p    = pix0 & 4095;

  const float4* xp = reinterpret_cast<const float4*>(
      x + (size_t)b * NC * HW + p);

  float4 v  = xp[0];                           // channel 0
  float4 mx = v;
  float4 sm = v;
#pragma unroll 4
  for (int c = 1; c < NC; ++c) {
    float4 u = xp[c * (HW / 4)];
    mx.x = fmaxf(mx.x, u.x); mx.y = fmaxf(mx.y, u.y);
    mx.z = fmaxf(mx.z, u.z); mx.w = fmaxf(mx.w, u.w);
    sm.x += u.x; sm.y += u.y; sm.z += u.z; sm.w += u.w;
  }
  const float s = 1.0f / 256.0f;
  float4 av = make_float4(sm.x * s, sm.y * s, sm.z * s, sm.w * s);

  float4* cp = reinterpret_cast<float4*>(comb + (size_t)b * 2 * HW + p);
  cp[0]        = mx;   // channel 0: max
  cp[HW / 4]   = av;   // channel 1: mean
}

// ---------------- Pass 2: 7x7 conv (WMMA) + sigmoid + gated copy ----------------
// Block = 256 threads = 8 waves; block tile = 8 rows x 64 cols of one image.
// LDS holds comb halo tile [2][14][72] (+3 pad each side) and padded weights.
// Wave w owns row h0+w. For each 32-wide half-row: two 16-pixel WMMA tiles.
// A(16x4) = weight taps replicated in every row -> after K-loop every lane's
// D[0] equals conv[lane%16]; lane L's gate = (L<16 ? acc0[0] : acc1[0]).
__global__ __launch_bounds__(256) void conv_gate_kernel(
    const float* __restrict__ x, const float* __restrict__ comb,
    const float* __restrict__ wv, const float* __restrict__ bias,
    float* __restrict__ out) {
  __shared__ float wlds[104];          // 98 taps + zero pad to K=100
  __shared__ float tile[2][14][72];    // [chan][y:-3..+10][x:-3..+66], pitch 72

  const int tid = threadIdx.x;
  const int b   = blockIdx.x >> 3;
  const int h0  = (blockIdx.x & 7) << 3;

  for (int e = tid; e < 104; e += 256) wlds[e] = (e < 98) ? wv[e] : 0.0f;
  for (int e = tid; e < 2 * 14 * 70; e += 256) {
    int i  = e / 980;
    int r  = e - i * 980;
    int y  = r / 70;
    int xx = r - y * 70;
    int gy = h0 + y - 3;
    int gx = xx - 3;
    float val = 0.0f;
    if ((unsigned)gy < 64u && (unsigned)gx < 64u)
      val = comb[(size_t)b * 2 * HW + i * HW + gy * WDIM + gx];
    tile[i][y][xx] = val;
  }
  __syncthreads();

  const int lane = tid & 31;
  const int wave = tid >> 5;       // row within the 8-row tile
  const int n    = lane & 15;      // WMMA column (pixel within 16-tile)
  const int half = lane >> 4;      // lane half selects K = 2*half + j
  const int row  = h0 + wave;
  const float bs = bias[0];

  for (int wseg = 0; wseg < 64; wseg += 32) {
    v8f acc0 = {};
    v8f acc1 = {};
#pragma unroll
    for (int k0 = 0; k0 < 100; k0 += 4) {
      int kA = k0 + 2 * half;          // K for VGPR component 0
      int kB = kA + 1;                 // K for VGPR component 1
      float a0 = wlds[kA];             // A row-replicated weights (broadcast)
      float a1 = wlds[kB];
      int kAe = kA > 97 ? 97 : kA;     // clamp pad taps (weight is 0 anyway)
      int kBe = kB > 97 ? 97 : kB;
      int iA = kAe / 49, rA = kAe - iA * 49, kyA = rA / 7, kxA = rA - kyA * 7;
      int iB = kBe / 49, rB = kBe - iB * 49, kyB = rB / 7, kxB = rB - kyB * 7;
      // B(4x16): lane -> (K = 2*half+j, N = n); im2col gather from LDS tile.
      float b00 = tile[iA][wave + kyA][wseg + n + kxA];
      float b01 = tile[iB][wave + kyB][wseg + n + kxB];
      float b10 = tile[iA][wave + kyA][wseg + 16 + n + kxA];
      float b11 = tile[iB][wave + kyB][wseg + 16 + n + kxB];
      v2f A  = {a0, a1};
      v2f B0 = {b00, b01};
      v2f B1 = {b10, b11};
      acc0 = __builtin_amdgcn_wmma_f32_16x16x4_f32(
          false, A, false, B0, (short)0, acc0, false, false);
      acc1 = __builtin_amdgcn_wmma_f32_16x16x4_f32(
          false, A, false, B1, (short)0, acc1, false, false);
    }
    float c0  = acc0[0] + bs;          // conv for pixel wseg + (lane%16)
    float c1  = acc1[0] + bs;          // conv for pixel wseg + 16 + (lane%16)
    float v   = (lane < 16) ? c0 : c1; // lane L's pixel = wseg + L
    float att = 1.0f / (1.0f + __expf(-v));

    // Gated copy: 32 lanes cover 32 consecutive w -> 128B coalesced per wave.
    int col = wseg + lane;
    const float* xp = x   + (size_t)b * NC * HW + row * WDIM + col;
    float*       op = out + (size_t)b * NC * HW + row * WDIM + col;
#pragma unroll 8
    for (int c = 0; c < NC; ++c) op[c * HW] = xp[c * HW] * att;
  }
}

extern "C" void kernel_launch(void* const* d_in, const int* in_sizes, int n_in,
                              void* d_out, int out_size, void* d_ws,
                              size_t ws_size, hipStream_t stream) {
  const float* x    = (const float*)d_in[0];   // [32,256,64,64]
  const float* wv   = (const float*)d_in[1];   // [1,2,7,7] OIHW flat
  const float* bias = (const float*)d_in[2];   // [1]
  float* out  = (float*)d_out;
  float* comb = (float*)d_ws;                  // [32][2][64][64] = 1 MiB

  // Pass 1: 32*64*64 pixels / 4 per thread / 256 per block = 128 blocks.
  reduce_maxmean_kernel<<<128, 256, 0, stream>>>(x, comb);
  // Pass 2: 32 images * 8 row-blocks = 256 blocks.
  conv_gate_kernel<<<256, 256, 0, stream>>>(x, comb, wv, bias, out);
}